// Solver_51694226375499
// MI455X (gfx1250) — compile-verified
//
#include <hip/hip_runtime.h>

typedef float v2f __attribute__((ext_vector_type(2)));
typedef float v8f __attribute__((ext_vector_type(8)));

#define N0 4096
#define N1 2048
#define N2 1024

// T = tridiag(1, -2, 1), the 1-D second-difference matrix (16x16)
__device__ __forceinline__ float Tval(int r, int c) {
  int d = r - c; if (d < 0) d = -d;
  return d == 0 ? -2.0f : (d == 1 ? 1.0f : 0.0f);
}

// residual after the first Jacobi sweep from u=0:  r = f + 0.2*lap(f) interior, f on ring
__device__ __forceinline__ float rval(const float* __restrict__ f, int n, int i, int j) {
  float c = f[i * n + j];
  if (i > 0 && i < n - 1 && j > 0 && j < n - 1)
    c += 0.2f * (f[(i - 1) * n + j] + f[(i + 1) * n + j] +
                 f[i * n + j - 1] + f[i * n + j + 1] - 4.0f * c);
  return c;
}

// fc = restrict( f - A(jacobi(0,f)) )   (2x2 average, stride 2)
__global__ __launch_bounds__(256) void k_res_restrict(const float* __restrict__ f, int n,
                                                      float* __restrict__ fc) {
  int nc = n >> 1;
  int J = blockIdx.x * 32 + threadIdx.x;
  int I = blockIdx.y * 8 + threadIdx.y;
  if (I >= nc || J >= nc) return;
  int i = I << 1, j = J << 1;
  float s = rval(f, n, i, j) + rval(f, n, i, j + 1) +
            rval(f, n, i + 1, j) + rval(f, n, i + 1, j + 1);
  fc[I * nc + J] = 0.25f * s;
}

__global__ __launch_bounds__(256) void k_scale(const float* __restrict__ src,
                                               float* __restrict__ dst, float c, int total) {
  int idx = blockIdx.x * 256 + threadIdx.x;
  if (idx < total) dst[idx] = c * src[idx];
}

// U = cf*f + prolong(uc); prolong is zero on the boundary ring.
// prolong = nearest-x2 upsample conv [.5 1 .5]^T [.5 1 .5]  -> 4-tap interp.
__device__ __forceinline__ float Uval(const float* __restrict__ f, const float* __restrict__ uc,
                                      int n, int i, int j, float cf) {
  float u = cf * f[i * n + j];
  if (i > 0 && i < n - 1 && j > 0 && j < n - 1) {
    int h = n >> 1;
    int I = i >> 1, J = j >> 1;
    int I2 = (i & 1) ? I + 1 : I - 1;
    int J2 = (j & 1) ? J + 1 : J - 1;
    float c00 = uc[I * h + J], c01 = uc[I * h + J2];
    float c10 = uc[I2 * h + J], c11 = uc[I2 * h + J2];
    u += 2.25f * c00 + 0.75f * (c01 + c10) + 0.25f * c11;
  }
  return u;
}

#define LDS_H 34
#define LDS_W 66
#define LDS_S 67

// out = jacobi( cf*f + prolong(uc), f )
// Block tile: 32 rows x 64 cols, 8 waves, each wave does a 16x16 subtile.
// lap(U) on each subtile computed on the matrix pipe: T*U + U*T via
// V_WMMA_F32_16X16X4_F32 (4 K-slices per product), halo terms added in VALU.
__global__ __launch_bounds__(256) void k_prolong_jacobi(const float* __restrict__ f,
                                                        const float* __restrict__ uc,
                                                        float* __restrict__ out,
                                                        int n, float cf, float cjac) {
  __shared__ float sU[LDS_H * LDS_S];
  int row0 = blockIdx.y * 32;
  int col0 = blockIdx.x * 64;
  int t = threadIdx.y * 32 + threadIdx.x;

  // cooperative fill of corrected field U (tile + 1-halo) into LDS
  for (int idx = t; idx < LDS_H * LDS_W; idx += 256) {
    int li = idx / LDS_W, lj = idx - li * LDS_W;
    int gi = row0 - 1 + li, gj = col0 - 1 + lj;
    float v = 0.0f;                       // out-of-domain halo only feeds masked ring outputs
    if (gi >= 0 && gi < n && gj >= 0 && gj < n) v = Uval(f, uc, n, gi, gj, cf);
    sU[li * LDS_S + lj] = v;
  }
  __syncthreads();

  int wave = threadIdx.y;
  int wr = (wave >> 2) * 16;   // subtile row origin in block tile (0 or 16)
  int wc = (wave & 3) * 16;    // subtile col origin (0..48)
  int lane = threadIdx.x;
  int hh = lane >> 4;          // lane half
  int l = lane & 15;

  // base points at subtile element (0,0) inside the haloed LDS tile
  const float* base = &sU[(wr + 1) * LDS_S + (wc + 1)];

  v8f acc = {};
  // T * U : A = T[:, kk..kk+3] (built from indices), B = U[kk..kk+3, :] (from LDS)
#pragma unroll
  for (int k4 = 0; k4 < 4; ++k4) {
    int kk = k4 * 4 + 2 * hh;  // A/B f32 layout: V0 = K even-half, V1 = K odd-half per lane half
    v2f a, b;
    a.x = Tval(l, kk);
    a.y = Tval(l, kk + 1);
    b.x = base[kk * LDS_S + l];
    b.y = base[(kk + 1) * LDS_S + l];
    acc = __builtin_amdgcn_wmma_f32_16x16x4_f32(false, a, false, b, (short)0, acc, false, false);
  }
  // U * T : A = U[:, kk..kk+3] (from LDS), B = T[kk..kk+3, :]
#pragma unroll
  for (int k4 = 0; k4 < 4; ++k4) {
    int kk = k4 * 4 + 2 * hh;
    v2f a, b;
    a.x = base[l * LDS_S + kk];
    a.y = base[l * LDS_S + kk + 1];
    b.x = Tval(kk, l);
    b.y = Tval(kk + 1, l);
    acc = __builtin_amdgcn_wmma_f32_16x16x4_f32(false, a, false, b, (short)0, acc, false, false);
  }

  // acc = lap(U) with zero-Dirichlet subtile edges; add the 4 halo edges, then Jacobi update
#pragma unroll
  for (int e = 0; e < 8; ++e) {
    int r = e + 8 * hh;   // C/D layout: VGPR e holds row e (lanes 0-15) / row e+8 (lanes 16-31)
    int c = l;
    float lap = acc[e];
    if (r == 0)  lap += base[-1 * LDS_S + c];
    if (r == 15) lap += base[16 * LDS_S + c];
    if (c == 0)  lap += base[r * LDS_S - 1];
    if (c == 15) lap += base[r * LDS_S + 16];
    int gi = row0 + wr + r;
    int gj = col0 + wc + c;
    float Ucen = base[r * LDS_S + c];
    float fv = f[gi * n + gj];
    bool interior = (gi > 0) && (gi < n - 1) && (gj > 0) && (gj < n - 1);
    float val = Ucen - cjac * fv;          // cjac = 0.2/n^2
    if (interior) val += 0.2f * lap;
    out[gi * n + gj] = val;
  }
}

// stage 1: per-block partial sums of |f| and |f - A(u)| (deterministic LDS tree)
__global__ __launch_bounds__(256) void k_resnorm_partial(const float* __restrict__ f,
                                                         const float* __restrict__ u,
                                                         int n, int shift,
                                                         float* __restrict__ part) {
  __shared__ float sf[256], sr[256];
  long total = (long)n * n;
  float n2 = (float)n * (float)n;
  float af = 0.0f, ar = 0.0f;
  long stride = (long)gridDim.x * 256;
  for (long idx = (long)blockIdx.x * 256 + threadIdx.x; idx < total; idx += stride) {
    int i = (int)(idx >> shift);
    int j = (int)(idx & (n - 1));
    float fv = f[idx];
    float r = fv;
    if (i > 0 && i < n - 1 && j > 0 && j < n - 1) {
      float lap = u[idx - n] + u[idx + n] + u[idx - 1] + u[idx + 1] - 4.0f * u[idx];
      r = fv - n2 * lap;
    }
    af += fabsf(fv);
    ar += fabsf(r);
  }
  sf[threadIdx.x] = af; sr[threadIdx.x] = ar;
  __syncthreads();
  for (int s = 128; s > 0; s >>= 1) {
    if ((int)threadIdx.x < s) {
      sf[threadIdx.x] += sf[threadIdx.x + s];
      sr[threadIdx.x] += sr[threadIdx.x + s];
    }
    __syncthreads();
  }
  if (threadIdx.x == 0) { part[2 * blockIdx.x] = sf[0]; part[2 * blockIdx.x + 1] = sr[0]; }
}

// stage 2: single-block deterministic final reduce, writes res = sum|r| / sum|f|
__global__ __launch_bounds__(256) void k_resnorm_final(const float* __restrict__ part,
                                                       int nblocks, float* __restrict__ res) {
  __shared__ float sf[256], sr[256];
  float af = 0.0f, ar = 0.0f;
  for (int i = threadIdx.x; i < nblocks; i += 256) { af += part[2 * i]; ar += part[2 * i + 1]; }
  sf[threadIdx.x] = af; sr[threadIdx.x] = ar;
  __syncthreads();
  for (int s = 128; s > 0; s >>= 1) {
    if ((int)threadIdx.x < s) {
      sf[threadIdx.x] += sf[threadIdx.x + s];
      sr[threadIdx.x] += sr[threadIdx.x + s];
    }
    __syncthreads();
  }
  if (threadIdx.x == 0) res[0] = sr[0] / sf[0];
}

extern "C" void kernel_launch(void* const* d_in, const int* in_sizes, int n_in,
                              void* d_out, int out_size, void* d_ws, size_t ws_size,
                              hipStream_t stream) {
  (void)in_sizes; (void)n_in; (void)out_size; (void)ws_size;
  const float* f0 = (const float*)d_in[0];
  // d_in[1] (R = const 0.25 2x2) and d_in[2] (P = fixed 3x3) are baked into the kernels.

  float* out_u = (float*)d_out;                       // 4096*4096 u, then 1 residual scalar
  float* ws = (float*)d_ws;
  float* f1 = ws;                                     // N1*N1
  float* f2 = f1 + (size_t)N1 * N1;                   // N2*N2
  float* u2 = f2 + (size_t)N2 * N2;                   // N2*N2
  float* u1 = u2 + (size_t)N2 * N2;                   // N1*N1
  float* part = u1 + (size_t)N1 * N1;                 // 2*RB

  dim3 blk(32, 8);

  // down-sweep: smooth(from 0) + residual + restrict, fused
  k_res_restrict<<<dim3(N1 / 32, N1 / 8), blk, 0, stream>>>(f0, N0, f1);
  k_res_restrict<<<dim3(N2 / 32, N2 / 8), blk, 0, stream>>>(f1, N1, f2);

  // coarsest smooth from 0:  u2 = -0.2/N2^2 * f2
  float c2 = -0.2f / ((float)N2 * (float)N2);
  k_scale<<<(N2 * N2) / 256, 256, 0, stream>>>(f2, u2, c2, N2 * N2);

  // up-sweep: prolong + correct + smooth, fused (WMMA Laplacian)
  float c1 = -0.2f / ((float)N1 * (float)N1);
  k_prolong_jacobi<<<dim3(N1 / 64, N1 / 32), blk, 0, stream>>>(f1, u2, u1, N1, c1,
                                                               0.2f / ((float)N1 * (float)N1));
  float c0 = -0.2f / ((float)N0 * (float)N0);
  k_prolong_jacobi<<<dim3(N0 / 64, N0 / 32), blk, 0, stream>>>(f0, u1, out_u, N0, c0,
                                                               0.2f / ((float)N0 * (float)N0));

  // residual norm: res = sum|f - A(u)| / sum|f|
  const int RB = 2048;
  k_resnorm_partial<<<RB, 256, 0, stream>>>(f0, out_u, N0, 12, part);
  k_resnorm_final<<<1, 256, 0, stream>>>(part, RB, out_u + (size_t)N0 * N0);
}